// GNN8_27410481283377
// MI455X (gfx1250) — compile-verified
//
#include <hip/hip_runtime.h>
#include <hip/hip_bf16.h>

typedef __attribute__((ext_vector_type(2))) float v2f;
typedef __attribute__((ext_vector_type(8))) float v8f;

#define Nn      500000
#define Kn      20
#define Vf      11
#define Ff      16
#define TILES   (Nn / 16)       // 31250, exact
#define BLOCKS  512
#define TPB     256
#define WPB     (TPB / 32)      // 8 waves/block (wave32)

// ---------------------------------------------------------------------------
// WMMA f32 16x16x4 : D = A(16x4) * B(4x16) + C(16x16), all fp32, wave32.
// ---------------------------------------------------------------------------
#if __has_builtin(__builtin_amdgcn_wmma_f32_16x16x4_f32)
#define HAS_WMMA_F32X4 1
#endif

static __device__ __forceinline__ v8f wmma4(v2f a, v2f b, v8f c) {
#ifdef HAS_WMMA_F32X4
    return __builtin_amdgcn_wmma_f32_16x16x4_f32(false, a, false, b,
                                                 (short)0, c, false, false);
#else
    asm volatile("v_wmma_f32_16x16x4_f32 %0, %1, %2, %0"
                 : "+v"(c) : "v"(a), "v"(b));
    return c;
#endif
}

// 8-byte paired load with only 4-byte alignment guarantee (rows are 44 B).
// Lowers to load <2 x float> align 4 -> global_load_b64 under unaligned mode,
// or two b32 loads otherwise (never worse than scalar).
static __device__ __forceinline__ v2f ldf2(const float* __restrict__ p) {
    v2f r;
    __builtin_memcpy(&r, p, sizeof(v2f));
    return r;
}

// ---------------------------------------------------------------------------
// 16-lane row reduction with DPP row_ror (VALU-only, no LDS, no waits).
// ---------------------------------------------------------------------------
template <int N>
static __device__ __forceinline__ float dpp_ror_add(float v) {
    int r = __builtin_amdgcn_update_dpp(0, __float_as_int(v),
                                        0x120 | N /*row_ror:N*/,
                                        0xF, 0xF, true);
    return v + __int_as_float(r);
}
static __device__ __forceinline__ float row_sum16(float s) {
    s = dpp_ror_add<1>(s);
    s = dpp_ror_add<2>(s);
    s = dpp_ror_add<4>(s);
    s = dpp_ror_add<8>(s);
    return s;   // every lane in each 16-lane row holds the row total
}

// Branch-free tanh via exp (scores here are small; clamp for safety).
static __device__ __forceinline__ float fast_tanh(float x) {
    x = fminf(fmaxf(x, -15.0f), 15.0f);
    float e = __expf(2.0f * x);
    return (e - 1.0f) / (e + 1.0f);
}

// B-matrix (4x16) operand for K-chunk kk of a [Vf x Ff] weight matrix.
static __device__ __forceinline__ v2f loadB(const float* __restrict__ W,
                                            int kk, bool hi, int m) {
    int r0 = kk * 4 + (hi ? 2 : 0);
    int r1 = r0 + 1;
    v2f r;
    r.x = (r0 < Vf) ? W[r0 * Ff + m] : 0.0f;
    r.y = (r1 < Vf) ? W[r1 * Ff + m] : 0.0f;
    return r;
}

// Fused attention-pool accumulation on a C-layout tile.
static __device__ __forceinline__ void att_acc(const v8f& x, float aw,
                                               float& num, float& den) {
#pragma unroll
    for (int r = 0; r < 8; ++r) {
        float s = row_sum16(x[r] * aw);      // row dot(att_w) per half
        float e = __expf(fast_tanh(s));
        den += e;
        num = fmaf(e, x[r], num);
    }
}

__global__ void __launch_bounds__(TPB) gnn_main(
    const float* __restrict__ vert,   // [N, 11]
    const float* __restrict__ Wc,     // [11, 16]
    const float* __restrict__ Wnh,    // [11, 16]
    const float* __restrict__ Wint,   // [11, 16]
    const float* __restrict__ bias,   // [16]
    const float* __restrict__ awInt,  // [16]
    const float* __restrict__ awNh,   // [16]
    const int*   __restrict__ nhIdx,  // [N, 20]
    const int*   __restrict__ intIdx, // [N, 20]
    float* __restrict__ partials)     // [BLOCKS, 34]
{
    const int  lane = threadIdx.x & 31;
    const int  wv   = threadIdx.x >> 5;
    const int  gw   = blockIdx.x * WPB + wv;
    const int  nw   = gridDim.x * WPB;
    const int  m    = lane & 15;
    const bool hi   = lane >= 16;
    const int  fb   = hi ? 2 : 0;     // base for pairs 0 and 1
    const int  o2   = hi ? 9 : 8;     // base for pair 2: (f8,f9) | (f9,f10)

    // Weight matrices resident in registers (3 K-chunks each)
    v2f bWc[3], bWn[3], bWi[3];
#pragma unroll
    for (int kk = 0; kk < 3; ++kk) {
        bWc[kk] = loadB(Wc,   kk, hi, m);
        bWn[kk] = loadB(Wnh,  kk, hi, m);
        bWi[kk] = loadB(Wint, kk, hi, m);
    }
    const float bv = bias[m];
    const float aI = awInt[m];
    const float aN = awNh[m];

    float numI = 0.0f, denI = 0.0f, numN = 0.0f, denN = 0.0f;

    for (int t = gw; t < TILES; t += nw) {
        const int node = t * 16 + m;
        const float* vrow = vert + (unsigned)node * (unsigned)Vf;

        // Center A operand: three paired loads, un-swizzled below.
        v2f pc0 = ldf2(vrow + fb);
        v2f pc1 = ldf2(vrow + fb + 4);
        v2f pc2 = ldf2(vrow + o2);

        // Neighbor-mean gathers. Index rows are 80 B / 16-B aligned -> int4.
        v2f pg0 = {}, pg1 = {}, pg2 = {};   // nh pairs
        v2f ph0 = {}, ph1 = {}, ph2 = {};   // int pairs
        const int4* __restrict__ ni4 = (const int4*)(nhIdx  + (long)node * Kn);
        const int4* __restrict__ ii4 = (const int4*)(intIdx + (long)node * Kn);
#pragma unroll
        for (int k4 = 0; k4 < Kn / 4; ++k4) {
            int4 an = ni4[k4];
            int4 ai = ii4[k4];
            int idx[8] = { an.x, an.y, an.z, an.w, ai.x, ai.y, ai.z, ai.w };
#pragma unroll
            for (int j = 0; j < 8; ++j) {
                const float* p = vert + (unsigned)idx[j] * (unsigned)Vf;
                v2f q0 = ldf2(p + fb);
                v2f q1 = ldf2(p + fb + 4);
                v2f q2 = ldf2(p + o2);
                if (j < 4) { pg0 += q0; pg1 += q1; pg2 += q2; }
                else       { ph0 += q0; ph1 += q1; ph2 += q2; }
            }
        }

        const float rk = 1.0f / (float)Kn;
        // Un-swizzle pair 2 (once per tile, not per row) + K=11 zero padding.
        v2f aC2, aG2, aH2;
        aC2.x = hi ? pc2.y : pc2.x;  aC2.y = hi ? 0.0f : pc2.y;
        aG2.x = (hi ? pg2.y : pg2.x) * rk;  aG2.y = hi ? 0.0f : pg2.y * rk;
        aH2.x = (hi ? ph2.y : ph2.x) * rk;  aH2.y = hi ? 0.0f : ph2.y * rk;

        v2f aC[3] = { pc0, pc1, aC2 };
        v2f aG[3] = { pg0 * rk, pg1 * rk, aG2 };
        v2f aH[3] = { ph0 * rk, ph1 * rk, aH2 };

        // center = vertices_tile @ Wc          (3x v_wmma_f32_16x16x4_f32)
        v8f ctr = {};
#pragma unroll
        for (int kk = 0; kk < 3; ++kk) ctr = wmma4(aC[kk], bWc[kk], ctr);

        // x_nh = relu(center + nh_mean @ Wn_nh + b)
        v8f xn;
#pragma unroll
        for (int r = 0; r < 8; ++r) xn[r] = ctr[r] + bv;
#pragma unroll
        for (int kk = 0; kk < 3; ++kk) xn = wmma4(aG[kk], bWn[kk], xn);
#pragma unroll
        for (int r = 0; r < 8; ++r) xn[r] = fmaxf(xn[r], 0.0f);

        // x_int = relu(center + int_mean @ Wn_int + b)
        v8f xi;
#pragma unroll
        for (int r = 0; r < 8; ++r) xi[r] = ctr[r] + bv;
#pragma unroll
        for (int kk = 0; kk < 3; ++kk) xi = wmma4(aH[kk], bWi[kk], xi);
#pragma unroll
        for (int r = 0; r < 8; ++r) xi[r] = fmaxf(xi[r], 0.0f);

        // Single-pass softmax pooling (tanh in [-1,1] -> no max pass needed).
        att_acc(xi, aI, numI, denI);
        att_acc(xn, aN, numN, denN);
    }

    // Combine the two half-wave partitions (rows 0-7 vs 8-15); once per wave.
    numI += __shfl_xor(numI, 16);
    numN += __shfl_xor(numN, 16);
    denI += __shfl_xor(denI, 16);
    denN += __shfl_xor(denN, 16);

    // Deterministic block reduction via LDS (no float atomics).
    __shared__ float lds[WPB][34];
    if (!hi) { lds[wv][m] = numI; lds[wv][16 + m] = numN; }
    if (lane == 0) { lds[wv][32] = denI; lds[wv][33] = denN; }
    __syncthreads();
    if (threadIdx.x < 34) {
        float s = 0.0f;
#pragma unroll
        for (int w = 0; w < WPB; ++w) s += lds[w][threadIdx.x];
        partials[blockIdx.x * 34 + threadIdx.x] = s;
    }
}

// Fixed-order final reduction + dense head. out = [r_int, r_nh] @ dense_W + b.
__global__ void gnn_final(const float* __restrict__ partials,
                          const float* __restrict__ dW,   // [32,1]
                          const float* __restrict__ db,   // [1]
                          float* __restrict__ out)
{
    __shared__ float tot[34];
    int t = threadIdx.x;
    if (t < 34) {
        float s = 0.0f;
        for (int b = 0; b < BLOCKS; ++b) s += partials[b * 34 + t];
        tot[t] = s;
    }
    __syncthreads();
    if (t == 0) {
        const float dI = tot[32], dN = tot[33];
        float acc = db[0];
#pragma unroll
        for (int f = 0; f < Ff; ++f) {
            acc += (tot[f]      / dI) * dW[f];        // r_int part
            acc += (tot[16 + f] / dN) * dW[Ff + f];   // r_nh part
        }
        out[0] = acc;
    }
}

extern "C" void kernel_launch(void* const* d_in, const int* in_sizes, int n_in,
                              void* d_out, int out_size, void* d_ws, size_t ws_size,
                              hipStream_t stream) {
    const float* vert  = (const float*)d_in[0];
    const float* Wc    = (const float*)d_in[1];
    const float* Wnh   = (const float*)d_in[2];
    const float* Wint  = (const float*)d_in[3];
    const float* b     = (const float*)d_in[4];
    const float* awInt = (const float*)d_in[5];
    const float* awNh  = (const float*)d_in[6];
    const float* dW    = (const float*)d_in[7];
    const float* db    = (const float*)d_in[8];
    const int*   nhIdx = (const int*)d_in[9];
    const int*   inIdx = (const int*)d_in[10];
    float* partials = (float*)d_ws;   // BLOCKS*34 floats, fully rewritten each call

    gnn_main<<<BLOCKS, TPB, 0, stream>>>(vert, Wc, Wnh, Wint, b, awInt, awNh,
                                         nhIdx, inIdx, partials);
    gnn_final<<<1, 64, 0, stream>>>(partials, dW, db, (float*)d_out);
}